// xlayer_65670049956569
// MI455X (gfx1250) — compile-verified
//
#include <hip/hip_runtime.h>

typedef __attribute__((ext_vector_type(2))) float v2f;
typedef __attribute__((ext_vector_type(4))) float v4f;
typedef __attribute__((ext_vector_type(8))) float v8f;

#define BATCH    4096
#define IN_DIM   384
#define XVAL_DIM 384
#define OUT_DIM  768
#define VCHUNKS  4
#define VPER     (XVAL_DIM / VCHUNKS)   // 96

// ---------------------------------------------------------------------------
// Stage 1: partial contraction  part[vc][i][o] = sum_{v in chunk vc} xval[v] * kernel[i][v][o]
// One block per (i, vc). 192 threads, each owns a float4 of the o dimension.
// kernel is streamed exactly once (452 MB) -> non-temporal loads, this is the
// 19.4 us HBM-bound stage that dominates the whole problem.
// ---------------------------------------------------------------------------
__global__ __launch_bounds__(192) void keff_partial(
    const float* __restrict__ xval,
    const float* __restrict__ kern,
    float* __restrict__ part)
{
    const int i  = blockIdx.x;           // 0..383  (input-dim index)
    const int vc = blockIdx.y;           // 0..3    (v chunk)
    const int o  = threadIdx.x * 4;      // 0..764

    const float* kp = kern + (size_t)i * (XVAL_DIM * OUT_DIM)
                           + (size_t)(vc * VPER) * OUT_DIM + o;
    const float* xv = xval + vc * VPER;

    v4f acc = {0.f, 0.f, 0.f, 0.f};
#pragma unroll 4
    for (int v = 0; v < VPER; ++v) {
        v4f kk = __builtin_nontemporal_load((const v4f*)kp);
        const float s = xv[v];           // uniform -> scalar load
        acc += s * kk;
        kp += OUT_DIM;
    }

    *(v4f*)(part + (size_t)(vc * IN_DIM + i) * OUT_DIM + o) = acc;
}

// ---------------------------------------------------------------------------
// Stage 1b: deterministic sequential reduce of the 4 partials -> w_eff[i][o]
// ---------------------------------------------------------------------------
__global__ __launch_bounds__(256) void keff_reduce(
    const float* __restrict__ part,
    float* __restrict__ weff)
{
    const int idx = blockIdx.x * 256 + threadIdx.x;   // float4 index
    const int n4  = (IN_DIM * OUT_DIM) / 4;           // 73728
    const v4f* p = (const v4f*)part;
    v4f s = p[idx];
    s += p[idx + n4];
    s += p[idx + 2 * n4];
    s += p[idx + 3 * n4];
    ((v4f*)weff)[idx] = s;
}

// ---------------------------------------------------------------------------
// Stage 2: out(4096x768) = x(4096x384) @ w_eff(384x768), exact f32 via
// V_WMMA_F32_16X16X4_F32. Block = 128 threads (4 waves, 2x2), block tile
// 128x128, wave tile 64x64 (16 accumulator fragments).
// A frag (16x4): lane<16 -> (M=lane, K=k0,k0+1); lane>=16 -> (M=lane-16, K=k0+2,k0+3)
// B frag (4x16): lane<16 -> (K=k0,k0+1,  N=lane); lane>=16 -> (K=k0+2,k0+3, N=lane-16)
// D frag: VGPR r holds row r + 8*(lane>=16), col = lane%16.
// ---------------------------------------------------------------------------
__global__ __launch_bounds__(128) void gemm_wmma_f32(
    const float* __restrict__ x,
    const float* __restrict__ w,
    float* __restrict__ out)
{
    const int lane = threadIdx.x & 31;
    const int wave = threadIdx.x >> 5;
    const int wm   = wave & 1;
    const int wn   = wave >> 1;
    const int m0   = blockIdx.x * 128 + wm * 64;
    const int n0   = blockIdx.y * 128 + wn * 64;
    const int lm   = lane & 15;
    const int kh   = lane >> 4;          // 0 or 1 (which K pair this lane holds)

    v8f acc[4][4];
#pragma unroll
    for (int mt = 0; mt < 4; ++mt)
#pragma unroll
        for (int nt = 0; nt < 4; ++nt)
            acc[mt][nt] = (v8f){0.f,0.f,0.f,0.f,0.f,0.f,0.f,0.f};

    for (int k0 = 0; k0 < IN_DIM; k0 += 4) {
        v2f a[4];
        v2f b[4];
#pragma unroll
        for (int mt = 0; mt < 4; ++mt) {
            const float* ap = x + (size_t)(m0 + mt * 16 + lm) * IN_DIM + (k0 + 2 * kh);
            a[mt] = *(const v2f*)ap;     // global_load_b64
        }
#pragma unroll
        for (int nt = 0; nt < 4; ++nt) {
            const float* bp = w + (size_t)(k0 + 2 * kh) * OUT_DIM + (n0 + nt * 16 + lm);
            b[nt].x = bp[0];
            b[nt].y = bp[OUT_DIM];
        }
#pragma unroll
        for (int mt = 0; mt < 4; ++mt)
#pragma unroll
            for (int nt = 0; nt < 4; ++nt)
                acc[mt][nt] = __builtin_amdgcn_wmma_f32_16x16x4_f32(
                    /*neg_a=*/false, a[mt],
                    /*neg_b=*/false, b[nt],
                    /*c_mod=*/(short)0, acc[mt][nt],
                    /*reuse_a=*/false, /*reuse_b=*/false);
    }

#pragma unroll
    for (int mt = 0; mt < 4; ++mt)
#pragma unroll
        for (int nt = 0; nt < 4; ++nt) {
            float* op = out + (size_t)(m0 + mt * 16 + 8 * kh) * OUT_DIM
                            + (n0 + nt * 16 + lm);
#pragma unroll
            for (int r = 0; r < 8; ++r)
                op[(size_t)r * OUT_DIM] = acc[mt][nt][r];
        }
}

// ---------------------------------------------------------------------------
extern "C" void kernel_launch(void* const* d_in, const int* in_sizes, int n_in,
                              void* d_out, int out_size, void* d_ws, size_t ws_size,
                              hipStream_t stream)
{
    const float* x    = (const float*)d_in[0];   // (4096, 384)
    const float* xval = (const float*)d_in[1];   // (384,)
    const float* kern = (const float*)d_in[2];   // (384, 384, 768)
    float*       out  = (float*)d_out;           // (4096, 768)

    float* part = (float*)d_ws;                               // 4 * 384*768 floats
    float* weff = part + (size_t)VCHUNKS * IN_DIM * OUT_DIM;  // 384*768 floats

    // Stage 1: stream the 452 MB kernel tensor once (HBM bound, ~19.4 us floor)
    keff_partial<<<dim3(IN_DIM, VCHUNKS), 192, 0, stream>>>(xval, kern, part);

    // Stage 1b: fold the 4 v-chunks deterministically
    keff_reduce<<<dim3((IN_DIM * OUT_DIM / 4) / 256), 256, 0, stream>>>(part, weff);

    // Stage 2: f32 WMMA GEMM (L2-resident operands, ~1 us)
    gemm_wmma_f32<<<dim3(BATCH / 128, OUT_DIM / 128), 128, 0, stream>>>(x, weff, out);
}